// SGC_32822140076408
// MI455X (gfx1250) — compile-verified
//
#include <hip/hip_runtime.h>
#include <math.h>

#define F_IN  128
#define F_OUT 64

typedef __attribute__((ext_vector_type(2))) float v2f;
typedef __attribute__((ext_vector_type(8))) float v8f;

// ---------------- utility ----------------
__global__ void fill_kernel(float* __restrict__ p, long n, float v) {
  long i = (long)blockIdx.x * blockDim.x + threadIdx.x;
  if (i < n) p[i] = v;
}

// deg[dst] += 1 per edge (deg pre-filled with 1.0 for self loops)
__global__ void degree_kernel(const int* __restrict__ ei, float* __restrict__ deg, int E) {
  int e = blockIdx.x * blockDim.x + threadIdx.x;
  if (e < E) atomicAdd(&deg[ei[E + e]], 1.0f);
}

__global__ void dinv_kernel(float* __restrict__ deg, int N) {
  int i = blockIdx.x * blockDim.x + threadIdx.x;
  if (i < N) deg[i] = rsqrtf(fmaxf(deg[i], 1.0f));
}

// One wave per edge (incl. implicit self loops): xout[dst] += dinv[s]*dinv[d] * xin[src]
__global__ void hop_kernel(const float* __restrict__ xin, float* __restrict__ xout,
                           const int* __restrict__ ei, const float* __restrict__ dinv,
                           int E, int N) {
  long tid = (long)blockIdx.x * blockDim.x + threadIdx.x;
  int wid  = (int)(tid >> 5);
  int lane = threadIdx.x & 31;
  if (wid >= E + N) return;
  int s, d;
  if (wid < E) { s = ei[wid]; d = ei[E + wid]; }
  else         { s = d = wid - E; }
  float nrm = dinv[s] * dinv[d];
  float4 v  = ((const float4*)(xin + (size_t)s * F_IN))[lane];
  float* od = xout + (size_t)d * F_IN + lane * 4;
  atomicAdd(od + 0, nrm * v.x);
  atomicAdd(od + 1, nrm * v.y);
  atomicAdd(od + 2, nrm * v.z);
  atomicAdd(od + 3, nrm * v.w);
}

// One wave per node: accumulate features + count into cluster buckets
__global__ void cluster_sum_kernel(const float* __restrict__ h, const int* __restrict__ cl,
                                   float* __restrict__ csum, float* __restrict__ ccnt, int N) {
  long tid = (long)blockIdx.x * blockDim.x + threadIdx.x;
  int wid  = (int)(tid >> 5);
  int lane = threadIdx.x & 31;
  if (wid >= N) return;
  int c = cl[wid];
  float4 v  = ((const float4*)(h + (size_t)wid * F_IN))[lane];
  float* cs = csum + (size_t)c * F_IN + lane * 4;
  atomicAdd(cs + 0, v.x);
  atomicAdd(cs + 1, v.y);
  atomicAdd(cs + 2, v.z);
  atomicAdd(cs + 3, v.w);
  if (lane == 0) atomicAdd(&ccnt[c], 1.0f);
}

__global__ void cluster_mean_kernel(float* __restrict__ csum, const float* __restrict__ ccnt, int C) {
  long i = (long)blockIdx.x * blockDim.x + threadIdx.x;
  if (i < (long)C * F_IN) csum[i] *= (1.0f / fmaxf(ccnt[i >> 7], 1.0f));
}

// One wave per 16-row stripe; computes all NCOL 16-wide column tiles.
// A:[M,128] row-major, B:[128, NCOL*16] row-major. Full-precision f32 WMMA.
// Tail rows are CLAMPED (not predicated) on load; store is guarded.
template <bool RELU, int NCOL>
__global__ void gemm_wmma_f32(const float* __restrict__ A, const float* __restrict__ B,
                              const float* __restrict__ bias, float* __restrict__ D,
                              int M) {
  const int Ncols = NCOL * 16;
  const int r0   = blockIdx.x << 4;
  const int lane = threadIdx.x & 31;
  const int half = lane >> 4;      // 0: K offsets {0,1}, 1: {2,3}
  const int l16  = lane & 15;

  int arow = r0 + l16;
  if (arow > M - 1) arow = M - 1;                       // clamp tail (store guarded)
  const float* Ap = A + (size_t)arow * F_IN + (half << 1);
  const float* Bp = B + (size_t)(half << 1) * Ncols + l16;

  v8f acc[NCOL];
#pragma unroll
  for (int c = 0; c < NCOL; ++c) acc[c] = (v8f){};

#pragma unroll
  for (int k0 = 0; k0 < F_IN; k0 += 4) {
    v2f a = *(const v2f*)(Ap + k0);                     // unconditional b64 load
    const float* Bk = Bp + (size_t)k0 * Ncols;
#pragma unroll
    for (int c = 0; c < NCOL; ++c) {
      v2f b;
      b.x = Bk[c * 16];                                 // row ka
      b.y = Bk[c * 16 + Ncols];                         // row ka+1
      acc[c] = __builtin_amdgcn_wmma_f32_16x16x4_f32(false, a, false, b, (short)0,
                                                     acc[c], false, false);
    }
  }

#pragma unroll
  for (int c = 0; c < NCOL; ++c) {
    float bs = bias[c * 16 + l16];
#pragma unroll
    for (int j = 0; j < 8; ++j) {
      int row = r0 + j + (half << 3);
      if (row < M) {
        float v = acc[c][j] + bs;
        if (RELU) v = fmaxf(v, 0.0f);
        D[(size_t)row * Ncols + c * 16 + l16] = v;
      }
    }
  }
}

// One wave per cluster row of 64: in-place log_softmax via wave32 shuffle reduction
__global__ void logsoftmax_kernel(float* __restrict__ h, int C) {
  long tid = (long)blockIdx.x * blockDim.x + threadIdx.x;
  int wid  = (int)(tid >> 5);
  int lane = threadIdx.x & 31;
  if (wid >= C) return;
  float* row = h + (size_t)wid * F_OUT;
  float v0 = row[lane], v1 = row[lane + 32];
  float m = fmaxf(v0, v1);
#pragma unroll
  for (int off = 16; off > 0; off >>= 1) m = fmaxf(m, __shfl_xor(m, off, 32));
  float s = expf(v0 - m) + expf(v1 - m);
#pragma unroll
  for (int off = 16; off > 0; off >>= 1) s += __shfl_xor(s, off, 32);
  float ls = logf(s);
  row[lane]      = v0 - m - ls;
  row[lane + 32] = v1 - m - ls;
}

// out[n] = lsm[cluster[n]]  (float4 granularity, 16 chunks per 64-wide row)
__global__ void gather_kernel(const float* __restrict__ lsm, const int* __restrict__ cl,
                              float* __restrict__ out, int N) {
  long t = (long)blockIdx.x * blockDim.x + threadIdx.x;
  if (t >= (long)N * 16) return;
  int n  = (int)(t >> 4);
  int ch = (int)(t & 15);
  int c  = cl[n];
  ((float4*)out)[(size_t)n * 16 + ch] = ((const float4*)lsm)[(size_t)c * 16 + ch];
}

extern "C" void kernel_launch(void* const* d_in, const int* in_sizes, int n_in,
                              void* d_out, int out_size, void* d_ws, size_t ws_size,
                              hipStream_t stream) {
  const float* x  = (const float*)d_in[0];
  const int*   ei = (const int*)d_in[1];
  const int*   cl = (const int*)d_in[2];
  const float* W1 = (const float*)d_in[3];
  const float* b1 = (const float*)d_in[4];
  const float* W2 = (const float*)d_in[5];
  const float* b2 = (const float*)d_in[6];
  float* out = (float*)d_out;

  const int N = in_sizes[2];
  const int E = in_sizes[1] / 2;
  const int C = N / 4;

  float* ws   = (float*)d_ws;
  float* deg  = ws;                               // N  (becomes dinv in place)
  float* bufA = deg + N;                          // N*F_IN
  float* bufB = bufA + (size_t)N * F_IN;          // N*F_IN
  float* csum = bufB + (size_t)N * F_IN;          // C*F_IN
  float* ccnt = csum + (size_t)C * F_IN;          // C
  float* h1   = ccnt + C;                         // C*F_IN
  float* h2   = h1 + (size_t)C * F_IN;            // C*F_OUT (log_softmax in place)

  const int TB = 256;
  auto nb = [](long n, int tb) { return (unsigned)((n + tb - 1) / tb); };

  // degrees + symmetric normalization
  fill_kernel<<<nb(N, TB), TB, 0, stream>>>(deg, N, 1.0f);
  degree_kernel<<<nb(E, TB), TB, 0, stream>>>(ei, deg, E);
  dinv_kernel<<<nb(N, TB), TB, 0, stream>>>(deg, N);

  // 2-hop propagation (scatter-add per edge, wave per edge)
  long nA = (long)N * F_IN;
  long hopThreads = (long)(E + N) * 32;
  fill_kernel<<<nb(nA, TB), TB, 0, stream>>>(bufA, nA, 0.0f);
  hop_kernel<<<nb(hopThreads, TB), TB, 0, stream>>>(x, bufA, ei, deg, E, N);
  fill_kernel<<<nb(nA, TB), TB, 0, stream>>>(bufB, nA, 0.0f);
  hop_kernel<<<nb(hopThreads, TB), TB, 0, stream>>>(bufA, bufB, ei, deg, E, N);

  // cluster mean compression
  long nC = (long)C * F_IN;
  fill_kernel<<<nb(nC, TB), TB, 0, stream>>>(csum, nC, 0.0f);
  fill_kernel<<<nb(C, TB), TB, 0, stream>>>(ccnt, C, 0.0f);
  cluster_sum_kernel<<<nb((long)N * 32, TB), TB, 0, stream>>>(bufB, cl, csum, ccnt, N);
  cluster_mean_kernel<<<nb(nC, TB), TB, 0, stream>>>(csum, ccnt, C);

  // MLP via full-precision f32 WMMA (wave per 16-row stripe, all col tiles)
  int rowTiles = (C + 15) / 16;
  gemm_wmma_f32<true,  F_IN  / 16><<<rowTiles, 32, 0, stream>>>(csum, W1, b1, h1, C);
  gemm_wmma_f32<false, F_OUT / 16><<<rowTiles, 32, 0, stream>>>(h1, W2, b2, h2, C);

  // per-cluster log_softmax, then scatter back to nodes
  logsoftmax_kernel<<<nb((long)C * 32, TB), TB, 0, stream>>>(h2, C);
  gather_kernel<<<nb((long)N * 16, TB), TB, 0, stream>>>(h2, cl, out, N);
}